// GAT_3788161155719
// MI455X (gfx1250) — compile-verified
//
#include <hip/hip_runtime.h>
#include <hip/hip_bf16.h>

// ---------------------------------------------------------------------------
// GAT (2-layer) for MI455X / gfx1250.
// GEMMs use V_WMMA_F32_16X16X4_F32 (full fp32 precision tensor op).
// Edge softmax/aggregation via encoded-uint atomicMax + f32 atomicAdd.
// ---------------------------------------------------------------------------

typedef __attribute__((ext_vector_type(2))) float v2f;
typedef __attribute__((ext_vector_type(8))) float v8f;

#define ENC_NEG_INF 0x007FFFFFu  // encf(-inf)

__device__ __forceinline__ unsigned encf(float f) {
  unsigned u = __float_as_uint(f);
  return (u & 0x80000000u) ? ~u : (u | 0x80000000u);
}
__device__ __forceinline__ float decf(unsigned u) {
  return (u & 0x80000000u) ? __uint_as_float(u ^ 0x80000000u)
                           : __uint_as_float(~u);
}
__device__ __forceinline__ float lrelu02(float v) { return v > 0.f ? v : 0.2f * v; }

__device__ __forceinline__ void edge_sd(const int* __restrict__ ei, int E, int e,
                                        int& s, int& d) {
  if (e < E) { s = ei[e]; d = ei[E + e]; }
  else       { s = e - E; d = e - E; }   // self-loop
}

// ---------------------------------------------------------------------------
// init kernels
// ---------------------------------------------------------------------------
__global__ void fill_f32(float* __restrict__ p, float v, int n) {
  int i = blockIdx.x * blockDim.x + threadIdx.x;
  if (i < n) p[i] = v;
}
__global__ void fill_u32(unsigned* __restrict__ p, unsigned v, int n) {
  int i = blockIdx.x * blockDim.x + threadIdx.x;
  if (i < n) p[i] = v;
}

// ---------------------------------------------------------------------------
// WMMA GEMM: D[nrows, NCOLS] = A[nrows, K] @ B[K, NCOLS]   (all fp32)
// One wave computes a 16-row tile; K-loop in steps of 4 using
// V_WMMA_F32_16X16X4_F32.  A-frag: lane m=lane&15, kh=lane>>4, vgpr v -> K=2kh+v.
// D-frag: vgpr r -> row r+8*kh, col lane&15.
// ---------------------------------------------------------------------------
template <int K, int NCOLS>
__global__ void gemm_wmma(const float* __restrict__ A, const float* __restrict__ B,
                          float* __restrict__ D, int nrows) {
  constexpr int NT = NCOLS / 16;
  const int wave = blockIdx.x * (blockDim.x >> 5) + (threadIdx.x >> 5);
  const int lane = threadIdx.x & 31;
  const int row0 = wave * 16;
  if (row0 >= nrows) return;            // wave-uniform: EXEC stays all-ones
  const int mn = lane & 15;
  const int kh = lane >> 4;

  v8f acc[NT] = {};                     // zero accumulators

  const float* arow = A + (size_t)(row0 + mn) * K;
#pragma unroll 4
  for (int k0 = 0; k0 < K; k0 += 4) {
    const int ka = k0 + 2 * kh;
    v2f a;
    a.x = arow[ka];
    a.y = arow[ka + 1];
#pragma unroll
    for (int t = 0; t < NT; ++t) {
      v2f b;
      b.x = B[(size_t)ka * NCOLS + t * 16 + mn];
      b.y = B[(size_t)(ka + 1) * NCOLS + t * 16 + mn];
      acc[t] = __builtin_amdgcn_wmma_f32_16x16x4_f32(
          false, a, false, b, 0, acc[t], false, false);
    }
  }

#pragma unroll
  for (int t = 0; t < NT; ++t)
#pragma unroll
    for (int r = 0; r < 8; ++r)
      D[(size_t)(row0 + r + 8 * kh) * NCOLS + t * 16 + mn] = acc[t][r];
}

// ---------------------------------------------------------------------------
// per-node attention logits: al_s[n,h] = sum_c h[n,h,c]*a_src[h,c]  (same dst)
// ---------------------------------------------------------------------------
__global__ void attn_logits(const float* __restrict__ h, const float* __restrict__ asrc,
                            const float* __restrict__ adst, float* __restrict__ als,
                            float* __restrict__ ald, int n, int heads, int ch) {
  int tid = blockIdx.x * blockDim.x + threadIdx.x;
  if (tid >= n * heads) return;
  int node = tid / heads;
  int hd = tid - node * heads;
  const float* hp = h + (size_t)(node * heads + hd) * ch;
  const float* sp = asrc + hd * ch;
  const float* dp = adst + hd * ch;
  float s = 0.f, d = 0.f;
  for (int c = 0; c < ch; ++c) {
    s += hp[c] * sp[c];
    d += hp[c] * dp[c];
  }
  als[tid] = s;
  ald[tid] = d;
}

// ---------------------------------------------------------------------------
// edge passes (scores recomputed each pass; one thread per edge-head)
// ---------------------------------------------------------------------------
__global__ void edge_max(const int* __restrict__ ei, int E, int N,
                         const float* __restrict__ als, const float* __restrict__ ald,
                         unsigned* __restrict__ menc, int heads) {
  int tid = blockIdx.x * blockDim.x + threadIdx.x;
  int total = (E + N) * heads;
  if (tid >= total) return;
  int e = tid / heads, hd = tid - e * heads;
  int s, d;
  edge_sd(ei, E, e, s, d);
  float sc = lrelu02(als[s * heads + hd] + ald[d * heads + hd]);
  atomicMax(&menc[d * heads + hd], encf(sc));
}

__global__ void edge_sumexp(const int* __restrict__ ei, int E, int N,
                            const float* __restrict__ als, const float* __restrict__ ald,
                            const unsigned* __restrict__ menc, float* __restrict__ z,
                            int heads) {
  int tid = blockIdx.x * blockDim.x + threadIdx.x;
  int total = (E + N) * heads;
  if (tid >= total) return;
  int e = tid / heads, hd = tid - e * heads;
  int s, d;
  edge_sd(ei, E, e, s, d);
  float sc = lrelu02(als[s * heads + hd] + ald[d * heads + hd]);
  float m = decf(menc[d * heads + hd]);
  atomicAdd(&z[d * heads + hd], expf(sc - m));
}

__global__ void edge_aggr(const int* __restrict__ ei, int E, int N,
                          const float* __restrict__ als, const float* __restrict__ ald,
                          const unsigned* __restrict__ menc, const float* __restrict__ z,
                          const float* __restrict__ h, float* __restrict__ agg,
                          int heads, int ch) {
  int tid = blockIdx.x * blockDim.x + threadIdx.x;
  int total = (E + N) * heads;
  if (tid >= total) return;
  int e = tid / heads, hd = tid - e * heads;
  int s, d;
  edge_sd(ei, E, e, s, d);
  float sc = lrelu02(als[s * heads + hd] + ald[d * heads + hd]);
  float m = decf(menc[d * heads + hd]);
  float alpha = expf(sc - m) / z[d * heads + hd];
  const float* hp = h + (size_t)(s * heads + hd) * ch;
  float* op = agg + (size_t)(d * heads + hd) * ch;
  for (int c = 0; c < ch; ++c) atomicAdd(&op[c], hp[c] * alpha);
}

// ---------------------------------------------------------------------------
// in-place bias + ELU (layer-1 output)
// ---------------------------------------------------------------------------
__global__ void bias_elu(float* __restrict__ agg, const float* __restrict__ b,
                         int n, int cols) {
  int tid = blockIdx.x * blockDim.x + threadIdx.x;
  if (tid >= n * cols) return;
  float v = agg[tid] + b[tid % cols];
  agg[tid] = v > 0.f ? v : expm1f(v);
}

// ---------------------------------------------------------------------------
// global sum pool: pooled[16] += sum over 1024-node chunk (LDS reduce)
// ---------------------------------------------------------------------------
__global__ void pool_sum(const float* __restrict__ agg2, float* __restrict__ pooled,
                         int n) {
  const int c = threadIdx.x & 15;
  const int ln = threadIdx.x >> 4;  // 0..15
  const int base = blockIdx.x * 1024;
  float acc = 0.f;
  for (int i = ln; i < 1024; i += 16) {
    int node = base + i;
    if (node < n) acc += agg2[(size_t)node * 16 + c];
  }
  __shared__ float sh[256];
  sh[threadIdx.x] = acc;
  __syncthreads();
  if (threadIdx.x < 16) {
    float s = 0.f;
    for (int j = 0; j < 16; ++j) s += sh[j * 16 + c];
    atomicAdd(&pooled[c], s);
  }
}

__global__ void final_softmax(const float* __restrict__ pooled,
                              const float* __restrict__ b2,
                              float* __restrict__ out, int n) {
  if (threadIdx.x == 0 && blockIdx.x == 0) {
    float v[16];
    float mx = -3.4e38f;
    for (int c = 0; c < 16; ++c) {
      v[c] = pooled[c] / (float)n + b2[c];
      mx = fmaxf(mx, v[c]);
    }
    float s = 0.f;
    for (int c = 0; c < 16; ++c) { v[c] = expf(v[c] - mx); s += v[c]; }
    for (int c = 0; c < 16; ++c) out[c] = v[c] / s;
  }
}

// ---------------------------------------------------------------------------
// host launcher
// ---------------------------------------------------------------------------
extern "C" void kernel_launch(void* const* d_in, const int* in_sizes, int n_in,
                              void* d_out, int out_size, void* d_ws, size_t ws_size,
                              hipStream_t stream) {
  const float* x   = (const float*)d_in[0];   // [N,128]
  const float* W1  = (const float*)d_in[1];   // [128,64]
  const float* as1 = (const float*)d_in[2];   // [8,8]
  const float* ad1 = (const float*)d_in[3];   // [8,8]
  const float* b1  = (const float*)d_in[4];   // [64]
  const float* W2  = (const float*)d_in[5];   // [64,16]
  const float* as2 = (const float*)d_in[6];   // [1,16]
  const float* ad2 = (const float*)d_in[7];   // [1,16]
  const float* b2  = (const float*)d_in[8];   // [16]
  const int*   ei  = (const int*)d_in[9];     // [2,E] flat
  const int N = in_sizes[10];                 // batch array length == #nodes
  const int E = in_sizes[9] / 2;
  const int ET = E + N;                       // with self-loops

  float* ws = (float*)d_ws;
  size_t o = 0;
  float* agg1 = ws + o; o += (size_t)N * 64;  // layer-1 aggregate -> h1 (elu)
  float* hf1  = ws + o; o += (size_t)N * 64;  // layer-1 features; recycled below
  float*    hf2  = hf1;                        // [N,16]
  float*    agg2 = hf1 + (size_t)N * 16;       // [N,16]
  float*    als2 = hf1 + (size_t)N * 32;       // [N]
  float*    ald2 = als2 + N;                   // [N]
  unsigned* m2   = (unsigned*)(ald2 + N);      // [N]
  float*    z2   = (float*)(m2 + N);           // [N]
  float* als1 = ws + o; o += (size_t)N * 8;
  float* ald1 = ws + o; o += (size_t)N * 8;
  unsigned* m1 = (unsigned*)(ws + o); o += (size_t)N * 8;
  float* z1   = ws + o; o += (size_t)N * 8;
  float* pooled = ws + o; o += 16;
  (void)ws_size; (void)n_in; (void)out_size;

  auto cdiv = [](int a, int b) { return (a + b - 1) / b; };
  const int B256 = 256;

  // ---- layer-1 init ----
  fill_f32<<<cdiv(N * 64, B256), B256, 0, stream>>>(agg1, 0.f, N * 64);
  fill_f32<<<cdiv(N * 8, B256), B256, 0, stream>>>(z1, 0.f, N * 8);
  fill_u32<<<cdiv(N * 8, B256), B256, 0, stream>>>(m1, ENC_NEG_INF, N * 8);
  fill_f32<<<1, 32, 0, stream>>>(pooled, 0.f, 16);

  // ---- layer 1: h = x @ W1 (WMMA fp32), logits, segment softmax, aggregate ----
  gemm_wmma<128, 64><<<cdiv(cdiv(N, 16), 8), 256, 0, stream>>>(x, W1, hf1, N);
  attn_logits<<<cdiv(N * 8, B256), B256, 0, stream>>>(hf1, as1, ad1, als1, ald1, N, 8, 8);
  edge_max<<<cdiv(ET * 8, B256), B256, 0, stream>>>(ei, E, N, als1, ald1, m1, 8);
  edge_sumexp<<<cdiv(ET * 8, B256), B256, 0, stream>>>(ei, E, N, als1, ald1, m1, z1, 8);
  edge_aggr<<<cdiv(ET * 8, B256), B256, 0, stream>>>(ei, E, N, als1, ald1, m1, z1,
                                                     hf1, agg1, 8, 8);
  bias_elu<<<cdiv(N * 64, B256), B256, 0, stream>>>(agg1, b1, N, 64);

  // ---- layer-2 init (recycled region; safe: stream-ordered after layer 1) ----
  fill_f32<<<cdiv(N * 16, B256), B256, 0, stream>>>(agg2, 0.f, N * 16);
  fill_f32<<<cdiv(N, B256), B256, 0, stream>>>(z2, 0.f, N);
  fill_u32<<<cdiv(N, B256), B256, 0, stream>>>(m2, ENC_NEG_INF, N);

  // ---- layer 2: h2 = elu(h1) @ W2 (WMMA fp32), softmax, aggregate ----
  gemm_wmma<64, 16><<<cdiv(cdiv(N, 16), 8), 256, 0, stream>>>(agg1, W2, hf2, N);
  attn_logits<<<cdiv(N, B256), B256, 0, stream>>>(hf2, as2, ad2, als2, ald2, N, 1, 16);
  edge_max<<<cdiv(ET, B256), B256, 0, stream>>>(ei, E, N, als2, ald2, m2, 1);
  edge_sumexp<<<cdiv(ET, B256), B256, 0, stream>>>(ei, E, N, als2, ald2, m2, z2, 1);
  edge_aggr<<<cdiv(ET, B256), B256, 0, stream>>>(ei, E, N, als2, ald2, m2, z2,
                                                 hf2, agg2, 1, 16);

  // ---- mean pool + bias + softmax ----
  pool_sum<<<cdiv(N, 1024), 256, 0, stream>>>(agg2, pooled, N);
  final_softmax<<<1, 32, 0, stream>>>(pooled, b2, (float*)d_out, N);
}